// PLP_3221225472195
// MI455X (gfx1250) — compile-verified
//
#include <hip/hip_runtime.h>
#include <math.h>

#define NN   100000
#define NN1  80000
#define DD   128
#define HH   64
#define CC   16
#define GG   3
#define EE   1000000
#define KK   20

typedef __attribute__((ext_vector_type(2))) float v2f;
typedef __attribute__((ext_vector_type(8))) float v8f;

// order-preserving float <-> uint transform for atomic segment-max
__device__ __forceinline__ unsigned f2ord(float f) {
    unsigned u = __float_as_uint(f);
    return (u & 0x80000000u) ? ~u : (u | 0x80000000u);
}
__device__ __forceinline__ float ord2f(unsigned u) {
    return __uint_as_float((u & 0x80000000u) ? (u & 0x7fffffffu) : ~u);
}

// ---------------- WMMA f32 GEMM: C[M, NT*16] = A[M,K] @ B[K, NT*16] (+bias) --------
// 256 threads = 8 waves per block; each wave owns a 16-row x (NT*16)-col strip:
// one A fragment feeds NT WMMAs. B (<=32KB) is staged in LDS once per block in
// k-pair-interleaved layout: element (k,n) lives at Bs[(k>>1)*Nc + n][k&1], so a
// lane's B fragment (two k-adjacent values, same n) is ONE contiguous
// ds_load_b64 straight into an even-aligned WMMA operand pair (no repacking).
// Lane layouts (V_WMMA_F32_16X16X4_F32, wave32):
//   A 16x4 : lane = m + 16*half holds K = 2*half + {0,1}
//   B 4x16 : lane = n + 16*half holds K = 2*half + {0,1}
//   C 16x16: lane = n + 16*half, VGPR r holds row m = 8*half + r
template <int NT>
__global__ __launch_bounds__(256)
void wmma_gemm_kernel(const float* __restrict__ A, const float* __restrict__ B,
                      const float* __restrict__ bias, float* __restrict__ Cout,
                      int M, int Kd) {
    constexpr int Nc = NT * 16;
    __shared__ v2f Bs[(128 / 2) * Nc];       // max K is 128 -> 64 k-pairs

    for (int idx = threadIdx.x; idx < Kd * Nc; idx += 256) {
        int k = idx / Nc;                    // Nc is constexpr -> shift
        int n = idx - k * Nc;
        ((float*)&Bs[(k >> 1) * Nc + n])[k & 1] = B[idx];
    }
    __syncthreads();

    const int wave = threadIdx.x >> 5;
    const int lane = threadIdx.x & 31;
    const int rowTile = blockIdx.x * 8 + wave;   // wave-uniform guard
    if (rowTile >= (M >> 4)) return;
    const int row0 = rowTile << 4;
    const int mn   = lane & 15;
    const int half = lane >> 4;
    const int kb   = half * 2;

    v8f acc[NT] = {};
    const float* ap = A + (size_t)(row0 + mn) * Kd + kb;   // 8B aligned (kb even)
    for (int k = 0; k < Kd; k += 4) {
        v2f a = *(const v2f*)(ap + k);                     // global_load_b64
        const v2f* brow = Bs + (size_t)((k + kb) >> 1) * Nc + mn;
#pragma unroll
        for (int t = 0; t < NT; ++t) {
            v2f b = brow[t * 16];                          // ds_load_b64
            acc[t] = __builtin_amdgcn_wmma_f32_16x16x4_f32(false, a, false, b,
                                                           (short)0, acc[t],
                                                           false, false);
        }
    }
#pragma unroll
    for (int t = 0; t < NT; ++t) {
        float bv = bias ? bias[t * 16 + mn] : 0.0f;
#pragma unroll
        for (int r = 0; r < 8; ++r) {
            Cout[(size_t)(row0 + half * 8 + r) * Nc + t * 16 + mn] = acc[t][r] + bv;
        }
    }
}

// ---------------- small utility kernels ----------------
__global__ void zero_kernel(float* __restrict__ p, int n) {
    int i = blockIdx.x * blockDim.x + threadIdx.x;
    if (i < n) p[i] = 0.0f;
}

__global__ void init_nodes_kernel(unsigned* __restrict__ mord, float* __restrict__ denom) {
    int i = blockIdx.x * blockDim.x + threadIdx.x;
    if (i >= NN) return;
    mord[i] = 0x007FFFFFu;   // f2ord(-inf)
    denom[i] = 0.0f;
}

// s[e] = dot64(fp[src], fp[dst]) / 8 ; atomic segment-max by dst
__global__ void edge_score_kernel(const float* __restrict__ fp,
                                  const int* __restrict__ src, const int* __restrict__ dst,
                                  float* __restrict__ es, unsigned* __restrict__ mord) {
    int e = blockIdx.x * blockDim.x + threadIdx.x;
    if (e >= EE) return;
    const float4* a = (const float4*)(fp + (size_t)src[e] * HH);
    const float4* b = (const float4*)(fp + (size_t)dst[e] * HH);
    float acc = 0.0f;
#pragma unroll
    for (int j = 0; j < HH / 4; ++j) {
        float4 x = a[j], y = b[j];
        acc += x.x * y.x + x.y * y.y + x.z * y.z + x.w * y.w;
    }
    float s = acc * 0.125f;                  // 1/sqrt(64)
    es[e] = s;
    atomicMax(&mord[dst[e]], f2ord(s));
}

// e[edge] = exp(s - m'); denom[dst] += e
__global__ void edge_exp_kernel(float* __restrict__ es, const int* __restrict__ dst,
                                const unsigned* __restrict__ mord, float* __restrict__ denom) {
    int e = blockIdx.x * blockDim.x + threadIdx.x;
    if (e >= EE) return;
    int d = dst[e];
    float m = ord2f(mord[d]);
    if (!isfinite(m)) m = 0.0f;              // matches where(isfinite(m), m, 0)
    float ev = __expf(es[e] - m);
    es[e] = ev;
    atomicAdd(&denom[d], ev);
}

__global__ void edge_att_kernel(float* __restrict__ es, const int* __restrict__ dst,
                                const float* __restrict__ denom) {
    int e = blockIdx.x * blockDim.x + threadIdx.x;
    if (e >= EE) return;
    es[e] = es[e] / (denom[dst[e]] + 1e-16f);
}

// out[dst, c4..c4+3] += att[e] * h[src, c4..c4+3]; one thread per edge*float4
__global__ void propagate_kernel(const float* __restrict__ att, const int* __restrict__ src,
                                 const int* __restrict__ dst, const float* __restrict__ h,
                                 float* __restrict__ out) {
    int tid = blockIdx.x * blockDim.x + threadIdx.x;
    if (tid >= EE * (CC / 4)) return;
    int e  = tid >> 2;
    int c4 = (tid & 3) * 4;
    float a = att[e];
    float4 hv = *(const float4*)(h + (size_t)src[e] * CC + c4);
    float* op = out + (size_t)dst[e] * CC + c4;
    atomicAdd(op + 0, a * hv.x);
    atomicAdd(op + 1, a * hv.y);
    atomicAdd(op + 2, a * hv.z);
    atomicAdd(op + 3, a * hv.w);
}

// h = h * (1-mask) + onehot * mask  (in place)
__global__ void mask_kernel(float* __restrict__ h, const unsigned char* __restrict__ mask,
                            const float* __restrict__ onehot) {
    int tid = blockIdx.x * blockDim.x + threadIdx.x;
    if (tid >= NN * CC) return;
    int i = tid >> 4;
    float m = mask[i] ? 1.0f : 0.0f;
    h[tid] = h[tid] * (1.0f - m) + onehot[tid] * m;
}

// z[i, j4..j4+3] = h0[i, .] + mean_k h1[nei[i,k], .]  (float4 lanes)
__global__ void ns_z_kernel(const float* __restrict__ h0, const float* __restrict__ h1,
                            const int* __restrict__ nei, float* __restrict__ z) {
    int tid = blockIdx.x * blockDim.x + threadIdx.x;
    if (tid >= NN * (HH / 4)) return;
    int i  = tid >> 4;
    int j4 = (tid & 15) * 4;
    const int* ni = nei + (size_t)i * KK;
    float4 s = make_float4(0.f, 0.f, 0.f, 0.f);
#pragma unroll
    for (int k = 0; k < KK; ++k) {
        float4 v = *(const float4*)(h1 + (size_t)ni[k] * HH + j4);
        s.x += v.x; s.y += v.y; s.z += v.z; s.w += v.w;
    }
    float4 a = *(const float4*)(h0 + (size_t)i * HH + j4);
    float4 r;
    r.x = a.x + s.x * (1.0f / KK);
    r.y = a.y + s.y * (1.0f / KK);
    r.z = a.z + s.z * (1.0f / KK);
    r.w = a.w + s.w * (1.0f / KK);
    *(float4*)(z + (size_t)i * HH + j4) = r;
}

// per-node G-softmax mix + sigmoid gate; writes logits and logits_lp
__global__ void combine_kernel(const float* __restrict__ hp,     // [G][N][C]
                               const float* __restrict__ att,    // [N][G]
                               const float* __restrict__ alpha,  // [N]
                               const float* __restrict__ ns,     // [N][C] (already in d_out)
                               float* __restrict__ out_logits, float* __restrict__ out_lp) {
    int tid = blockIdx.x * blockDim.x + threadIdx.x;
    if (tid >= NN * CC) return;
    int i = tid >> 4;
    float a0 = att[i * 3 + 0], a1 = att[i * 3 + 1], a2 = att[i * 3 + 2];
    float mx = fmaxf(a0, fmaxf(a1, a2));
    float e0 = __expf(a0 - mx), e1 = __expf(a1 - mx), e2 = __expf(a2 - mx);
    float inv = 1.0f / (e0 + e1 + e2);
    float lp = (e0 * hp[(size_t)0 * NN * CC + tid] +
                e1 * hp[(size_t)1 * NN * CC + tid] +
                e2 * hp[(size_t)2 * NN * CC + tid]) * inv;
    float al = alpha[i];
    float g  = 1.0f / (1.0f + __expf(-al));  // sigmoid(alpha)
    float g2 = 1.0f / (1.0f + __expf(al));   // sigmoid(-alpha)
    out_lp[tid] = lp;
    out_logits[tid] = g * lp + g2 * ns[tid];
}

extern "C" void kernel_launch(void* const* d_in, const int* in_sizes, int n_in,
                              void* d_out, int out_size, void* d_ws, size_t ws_size,
                              hipStream_t stream) {
    (void)in_sizes; (void)n_in; (void)out_size; (void)ws_size;

    const float* feats0     = (const float*)d_in[0];
    const float* feats1     = (const float*)d_in[1];
    const float* W0         = (const float*)d_in[2];
    const float* b0         = (const float*)d_in[3];
    const float* W1         = (const float*)d_in[4];
    const float* b1         = (const float*)d_in[5];
    const float* Wp         = (const float*)d_in[6];
    const float* Wcls       = (const float*)d_in[7];
    const float* bcls       = (const float*)d_in[8];
    const float* alpha      = (const float*)d_in[9];
    const float* attention  = (const float*)d_in[10];
    const float* labels_oh  = (const float*)d_in[11];
    const float* label_init = (const float*)d_in[12];
    const unsigned char* mask = (const unsigned char*)d_in[13];
    const int*   nei        = (const int*)d_in[14];

    // workspace partition (floats)
    float* ws = (float*)d_ws;
    size_t off = 0;
    float* fp = ws + off;  off += (size_t)NN  * HH;
    float* h0 = ws + off;  off += (size_t)NN  * HH;
    float* h1 = ws + off;  off += (size_t)NN1 * HH;
    float* z  = ws + off;  off += (size_t)NN  * HH;
    float* hp = ws + off;  off += (size_t)GG * NN * CC;
    float* hA = ws + off;  off += (size_t)NN * CC;
    float* hB = ws + off;  off += (size_t)NN * CC;
    float* es = ws + off;  off += (size_t)EE;
    unsigned* mord = (unsigned*)(ws + off); off += (size_t)NN;
    float* denom = ws + off; off += (size_t)NN;

    float* out_logits = (float*)d_out;
    float* out_lp = out_logits + (size_t)NN * CC;
    float* out_ns = out_logits + 2 * (size_t)NN * CC;

    // dense projections (WMMA path, B staged in LDS, pair-interleaved)
    const int gb0 = (NN / 16 + 7) / 8;        // 6250 row tiles -> 782 blocks
    const int gb1 = (NN1 / 16 + 7) / 8;       // 5000 row tiles -> 625 blocks
    wmma_gemm_kernel<4><<<gb0, 256, 0, stream>>>(feats0, Wp, nullptr, fp, NN, DD);
    wmma_gemm_kernel<4><<<gb0, 256, 0, stream>>>(feats0, W0, b0, h0, NN, DD);
    wmma_gemm_kernel<4><<<gb1, 256, 0, stream>>>(feats1, W1, b1, h1, NN1, DD);

    // NS branch: neighbor mean + classifier GEMM straight into d_out slice
    ns_z_kernel<<<(NN * (HH / 4) + 255) / 256, 256, 0, stream>>>(h0, h1, nei, z);
    wmma_gemm_kernel<1><<<gb0, 256, 0, stream>>>(z, Wcls, bcls, out_ns, NN, HH);

    // label propagation over 3 metapath graphs; attention computed once/graph
    const int eb  = (EE + 255) / 256;
    const int nb  = (NN + 255) / 256;
    const int ncb = (NN * CC + 255) / 256;
    const int epb = (EE * (CC / 4) + 255) / 256;
    for (int g = 0; g < GG; ++g) {
        const int* src = (const int*)d_in[15 + 2 * g];
        const int* dst = (const int*)d_in[16 + 2 * g];
        init_nodes_kernel<<<nb, 256, 0, stream>>>(mord, denom);
        edge_score_kernel<<<eb, 256, 0, stream>>>(fp, src, dst, es, mord);
        edge_exp_kernel<<<eb, 256, 0, stream>>>(es, dst, mord, denom);
        edge_att_kernel<<<eb, 256, 0, stream>>>(es, dst, denom);

        const float* hcur = label_init;
        for (int l = 0; l < 3; ++l) {
            float* hnext = (l == 2) ? (hp + (size_t)g * NN * CC) : ((l & 1) ? hB : hA);
            zero_kernel<<<ncb, 256, 0, stream>>>(hnext, NN * CC);
            propagate_kernel<<<epb, 256, 0, stream>>>(es, src, dst, hcur, hnext);
            mask_kernel<<<ncb, 256, 0, stream>>>(hnext, mask, labels_oh);
            hcur = hnext;
        }
    }

    combine_kernel<<<ncb, 256, 0, stream>>>(hp, attention, alpha, out_ns,
                                            out_logits, out_lp);
}